// PytorchCreator_29884382445677
// MI455X (gfx1250) — compile-verified
//
#include <hip/hip_runtime.h>
#include <hip/hip_bf16.h>

typedef __attribute__((ext_vector_type(2))) float v2f;
typedef __attribute__((ext_vector_type(8))) float v8f;

#define L_EDGE 3
#define NNODES 64
#define KFAN   16
#define DDIM   256
#define KD     (KFAN * DDIM)   // 4096 flattened reduction depth
#define KQ     (KD / 4)        // 1024 per reduction quarter

// ---------------------------------------------------------------------------
// h0 = sigmoid(x0)
// ---------------------------------------------------------------------------
__global__ __launch_bounds__(256) void sigmoid_in_kernel(
    const float* __restrict__ x, float* __restrict__ h, int n) {
  int i = blockIdx.x * blockDim.x + threadIdx.x;
  if (i < n) h[i] = 1.0f / (1.0f + __expf(-x[i]));
}

// ---------------------------------------------------------------------------
// zbuf[n, e] = sum_k b[n, k, e]   (initializes accumulation target)
// grid: 64 blocks (one per node), 256 threads (one per e)
// ---------------------------------------------------------------------------
__global__ __launch_bounds__(256) void bias_init_kernel(
    const float* __restrict__ b, float* __restrict__ zbuf) {
  const int node = blockIdx.x;
  const int e    = threadIdx.x;
  float s = 0.0f;
#pragma unroll
  for (int k = 0; k < KFAN; ++k)
    s += b[(node * KFAN + k) * DDIM + e];
  zbuf[node * DDIM + e] = s;
}

// ---------------------------------------------------------------------------
// Partial GEMV via V_WMMA_F32_16X16X4_F32.
// blockIdx.x encodes (node, e-half, kd-quarter):  node = bx>>3,
// ehalf = (bx>>2)&1, kq = bx&3.  8 waves/block, each wave owns a 16-wide
// e tile; each block reduces one 1024-deep kd quarter and atomically adds
// its partial into zbuf.
// ---------------------------------------------------------------------------
__global__ __launch_bounds__(256, 4) void layer_partial_kernel(
    const float* __restrict__ hprev,   // [N, D]
    const float* __restrict__ W,       // [N, K, D, D] (this layer)
    const int*   __restrict__ pidx,    // [N, K]       (this layer)
    float*       __restrict__ zbuf)    // [N, D] accumulation target
{
  alignas(16) __shared__ float g[KQ];  // gathered activations, this quarter

  const int bx    = blockIdx.x;
  const int node  = bx >> 3;
  const int ehalf = (bx >> 2) & 1;
  const int kq    = bx & 3;
  const int tid   = threadIdx.x;

  const int kd_begin = kq << 10;       // quarter start in flattened (k,d)

  // Cooperative gather: g[i] = hprev[parent(k), d] for kd = kd_begin + i
#pragma unroll
  for (int i = tid; i < KQ; i += 256) {
    const int kd = kd_begin + i;
    const int k  = kd >> 8;
    const int d  = kd & 255;
    const int p  = pidx[node * KFAN + k];
    g[i] = hprev[p * DDIM + d];
  }
  __syncthreads();

  const int lane   = tid & 31;
  const int wave   = tid >> 5;
  const int e0     = ehalf * 128 + wave * 16;
  const int col    = lane & 15;
  const int rowoff = (lane >> 4) << 1;            // 0 (lanes 0-15) or 2

  // Per-lane streaming pointer into this node's weight slab:
  // element (kd + rowoff, e0 + col), row stride DDIM floats.
  const float* __restrict__ wptr =
      W + (size_t)node * KFAN * DDIM * DDIM + (size_t)rowoff * DDIM + e0 + col;

  v8f c = {};                                     // fp32 accumulator tile

  for (int kdo = kd_begin; kdo < kd_begin + KQ; kdo += 16) {
    // Prime L2 64 rows (64 KB) ahead; wrap inside the node's 4 MB slab.
    __builtin_prefetch(wptr + (size_t)((kdo + 64) & (KD - 1)) * DDIM, 0, 1);
#pragma unroll
    for (int u = 0; u < 16; u += 4) {
      const int kd = kdo + u;
      // A fragment: broadcast GEMV row -> ds_load_b64 of two consecutive g's
      const float2 ga = *reinterpret_cast<const float2*>(&g[(kd & (KQ - 1)) + rowoff]);
      v2f a;  a.x = ga.x;  a.y = ga.y;
      // B fragment: rows (kd+rowoff, kd+rowoff+1), column (e0+col)
      v2f bm;
      bm.x = wptr[(size_t)kd * DDIM];
      bm.y = wptr[(size_t)kd * DDIM + DDIM];
      c = __builtin_amdgcn_wmma_f32_16x16x4_f32(
          /*neg_a=*/false, a, /*neg_b=*/false, bm,
          /*c_mod=*/(short)0, c, /*reuse_a=*/false, /*reuse_b=*/false);
    }
  }

  // D layout: VGPR0, lanes 0-15 hold (M=0, N=0..15) -> z[e0 + col].
  if (lane < 16) {
    const float partial = c[0];
    __hip_atomic_fetch_add(&zbuf[node * DDIM + e0 + col], partial,
                           __ATOMIC_RELAXED, __HIP_MEMORY_SCOPE_AGENT);
  }
}

// ---------------------------------------------------------------------------
// hnext = sigmoid(zbuf)
// ---------------------------------------------------------------------------
__global__ __launch_bounds__(256) void activate_kernel(
    const float* __restrict__ zbuf, float* __restrict__ hnext, int n) {
  int i = blockIdx.x * blockDim.x + threadIdx.x;
  if (i < n) hnext[i] = 1.0f / (1.0f + __expf(-zbuf[i]));
}

// ---------------------------------------------------------------------------
extern "C" void kernel_launch(void* const* d_in, const int* in_sizes, int n_in,
                              void* d_out, int out_size, void* d_ws, size_t ws_size,
                              hipStream_t stream) {
  const float* x0   = (const float*)d_in[0];
  const float* W    = (const float*)d_in[1];
  const float* b    = (const float*)d_in[2];
  const int*   pidx = (const int*)d_in[3];
  float* out = (float*)d_out;

  const int ND = NNODES * DDIM;                   // 16384
  float* hA   = (float*)d_ws;                     // ping
  float* hB   = hA + ND;                          // pong
  float* zbuf = hB + ND;                          // pre-activation accumulator

  sigmoid_in_kernel<<<(ND + 255) / 256, 256, 0, stream>>>(x0, hA, ND);

  const size_t Wl = (size_t)NNODES * KFAN * DDIM * DDIM;  // per-layer W elems
  const size_t bl = (size_t)NNODES * KFAN * DDIM;
  const size_t pl = (size_t)NNODES * KFAN;

  float* hbuf[L_EDGE + 1] = { hA, hB, hA, out };
  for (int l = 0; l < L_EDGE; ++l) {
    bias_init_kernel<<<NNODES, 256, 0, stream>>>(b + (size_t)l * bl, zbuf);
    layer_partial_kernel<<<NNODES * 8, 256, 0, stream>>>(
        hbuf[l], W + (size_t)l * Wl, pidx + (size_t)l * pl, zbuf);
    activate_kernel<<<(ND + 255) / 256, 256, 0, stream>>>(zbuf, hbuf[l + 1], ND);
  }
}